// MultiHeadedRelativeAttention_67903432950269
// MI455X (gfx1250) — compile-verified
//
#include <hip/hip_runtime.h>
#include <stdint.h>

// ---------------------------------------------------------------------------
// MultiHeadedRelativeAttention for MI455X (gfx1250).
//  * full-f32 matrix math via V_WMMA_F32_16X16X4_F32 (problem is L2/latency
//    bound, ~3 GFLOP total, so no reason to drop precision)
//  * rel-pos term restructured: P[b,h,i,t] = q . emb[t]  (GEMM), then
//    QR[b,h,i,j] = P[b,h,i,dists[b,i,j]] (scalar gather) -- avoids the 537MB
//    emb[dists] materialization of the naive einsum
//  * CDNA5 async data movement: GLOBAL_LOAD_ASYNC_TO_LDS_B128 stages the
//    GEMM weight tile and the attention V panel into LDS, overlapped with
//    WMMA compute, synchronized with S_WAIT_ASYNCCNT.
// ---------------------------------------------------------------------------

#define HH      8      // heads
#define DM      512    // d_model
#define DK      64     // d_k
#define SS      256    // sequence length
#define BB      4      // batch
#define MAXT    128    // rel-pos table size

#define BS_LDB  520    // padded LDS row stride (floats) for GEMM B tile
#define VS_LDB  68     // padded LDS row stride (floats) for V panel

typedef __attribute__((ext_vector_type(2))) float v2f;
typedef __attribute__((ext_vector_type(8))) float v8f;

// ---- CDNA5 async copy: global -> LDS, 16B per lane, tracked by ASYNCcnt ---
__device__ __forceinline__ void async_copy_b128(uint32_t lds_off_bytes,
                                                const float* gptr) {
#if defined(__gfx1250__)
  asm volatile("global_load_async_to_lds_b128 %0, %1, off"
               :
               : "v"(lds_off_bytes), "v"((uint64_t)(uintptr_t)gptr)
               : "memory");
#endif
}

__device__ __forceinline__ void async_wait_all() {
#if defined(__gfx1250__)
#if __has_builtin(__builtin_amdgcn_s_wait_asynccnt)
  __builtin_amdgcn_s_wait_asynccnt(0);
#else
  asm volatile("s_wait_asynccnt 0x0" ::: "memory");
#endif
#endif
}

// LDS byte offset of a __shared__ pointer: generic LDS addresses carry the
// wave-relative offset in addr[31:0] (ISA 10.2 aperture mapping).
__device__ __forceinline__ uint32_t lds_offset(const void* p) {
  return (uint32_t)(uintptr_t)p;
}

// ---- wave-level 16x16 tile GEMM, C += A(16xK) * B^T(16xK)  (both k-major) --
// f32 WMMA fragment layout (ISA 7.12.2):
//   A 16x4 : lanes 0-15 row M=lane, VGPR{0,1}=K{0,1}; lanes 16-31 K{2,3}
//   B 4x16 : symmetric (lane = column N, VGPR pair = two K rows)
//   C 16x16: VGPR r -> (M=r, N=lane) for lanes 0-15, (M=r+8, N=lane-16)
__device__ __forceinline__ v8f wmma_xt(const float* __restrict__ a_tile, int lda,
                                       const float* __restrict__ b_tile, int ldb,
                                       int kdim, v8f acc) {
  const int lane = threadIdx.x & 31;
  const int l    = lane & 15;
  const int half = lane >> 4;
  const float* ap = a_tile + (size_t)l * lda + 2 * half;
  const float* bp = b_tile + (size_t)l * ldb + 2 * half;
#pragma unroll 8
  for (int k = 0; k < kdim; k += 4) {
    v2f av, bv;
    av.x = ap[k];
    av.y = ap[k + 1];
    bv.x = bp[k];
    bv.y = bp[k + 1];
    acc = __builtin_amdgcn_wmma_f32_16x16x4_f32(false, av, false, bv,
                                                (short)0, acc, false, false);
  }
  return acc;
}

__device__ __forceinline__ void store_tile(float* __restrict__ c_tile, int ldc,
                                           v8f acc, const float* __restrict__ bias_n0) {
  const int lane = threadIdx.x & 31;
  const int l    = lane & 15;
  const int half = lane >> 4;
  const float bv = bias_n0 ? bias_n0[l] : 0.0f;
#pragma unroll
  for (int r = 0; r < 8; ++r)
    c_tile[(size_t)(r + 8 * half) * ldc + l] = acc[r] + bv;
}

// ---- C[m,n] = sum_k A[m,k]*Bm[n,k] + bias[n],  K fixed at 512 -------------
// Block = 4 waves stacked on M (64x16 tile). The 16x512 B tile is shared by
// all 4 waves: stage it once into LDS with coalesced async b128 copies, then
// the inner loop reads A from global (b64) and B from LDS (conflict-free
// ds_load_b64 thanks to the 520-float row pad: bank = (8*l + k) % 64).
__global__ void __launch_bounds__(128)
gemm_xt_bias(const float* __restrict__ A, const float* __restrict__ Bm,
             const float* __restrict__ bias, float* __restrict__ C,
             int lda, int ldb, int ldc) {
  __shared__ float Bs[16 * BS_LDB];          // ~33 KB
  const int wave = threadIdx.x >> 5;
  const int lane = threadIdx.x & 31;
  const int l = lane & 15, half = lane >> 4;
  const int m0 = blockIdx.y * 64 + wave * 16;
  const int n0 = blockIdx.x * 16;

  // async stage B tile: 16 rows x 512 floats = 2048 b128 transfers
  {
    const uint32_t base = lds_offset(Bs);
#pragma unroll
    for (int it = 0; it < 16; ++it) {
      const int idx = it * 128 + (int)threadIdx.x;
      const int n = idx >> 7;                // row 0..15
      const int c = idx & 127;               // 16B chunk 0..127
      async_copy_b128(base + (uint32_t)(n * BS_LDB + c * 4) * 4u,
                      Bm + (size_t)(n0 + n) * ldb + c * 4);
    }
  }
  async_wait_all();
  __syncthreads();

  const float* ap = A + (size_t)(m0 + l) * lda + 2 * half;
  const float* bp = Bs + l * BS_LDB + 2 * half;
  v8f acc = {};
#pragma unroll 8
  for (int k = 0; k < 512; k += 4) {
    v2f av, bv;
    av.x = ap[k];
    av.y = ap[k + 1];
    bv.x = bp[k];
    bv.y = bp[k + 1];
    acc = __builtin_amdgcn_wmma_f32_16x16x4_f32(false, av, false, bv,
                                                (short)0, acc, false, false);
  }
  store_tile(C + (size_t)m0 * ldc + n0, ldc, acc, bias + n0);
}

// ---- P[b,h,i,t] = sum_d Q[b,i,h*64+d] * emb[t,h*64+d] ---------------------
__global__ void __launch_bounds__(128)
p_gemm(const float* __restrict__ Q, const float* __restrict__ emb,
       float* __restrict__ P) {
  const int bh = blockIdx.z;                 // b*H + h
  const int b = bh >> 3, h = bh & 7;
  const int wave = threadIdx.x >> 5;
  const int m0 = blockIdx.y * 64 + wave * 16;
  const int n0 = blockIdx.x * 16;
  const float* a_tile = Q + ((size_t)(b * SS + m0)) * DM + h * DK;
  const float* b_tile = emb + (size_t)n0 * DM + h * DK;
  v8f acc = {};
  acc = wmma_xt(a_tile, DM, b_tile, DM, DK, acc);
  store_tile(P + ((size_t)bh * SS + m0) * MAXT + n0, MAXT, acc, nullptr);
}

// ---- attention: per (b,h, 16-row i block) ---------------------------------
__global__ void __launch_bounds__(128)
attention_kernel(const float* __restrict__ Q, const float* __restrict__ K,
                 const float* __restrict__ V, const float* __restrict__ P,
                 const int* __restrict__ dists, const int* __restrict__ mask,
                 float* __restrict__ X) {
  __shared__ float sc[16][SS];               // 16 KB score/prob tile
  __shared__ float Vs[SS * VS_LDB];          // 68 KB staged V_h panel
  const int bh = blockIdx.y;
  const int b = bh >> 3, h = bh & 7;
  const int i0 = blockIdx.x * 16;
  const int wave = threadIdx.x >> 5;
  const int lane = threadIdx.x & 31;
  const int l = lane & 15, half = lane >> 4;

  // Kick off async staging of the whole V_h panel (256 x 64) into LDS.
  // 4096 b128 transfers -> 32 async instructions per wave (ASYNCcnt <= 32).
  // This overlaps with the QK^T WMMA phase and the softmax below.
  {
    const float* vsrc = V + (size_t)b * SS * DM + h * DK;
    const uint32_t base = lds_offset(Vs);
#pragma unroll
    for (int it = 0; it < 32; ++it) {
      const int idx = it * 128 + (int)threadIdx.x;
      const int j = idx >> 4;                // row 0..255
      const int dd = idx & 15;               // 16B chunk 0..15
      async_copy_b128(base + (uint32_t)(j * VS_LDB + dd * 4) * 4u,
                      vsrc + (size_t)j * DM + dd * 4);
    }
  }

  // Phase 1: scores = Q_h (16x64) . K_h^T  -> LDS  (4 waves x 4 j-tiles)
  const float* Qt = Q + ((size_t)(b * SS + i0)) * DM + h * DK;
#pragma unroll
  for (int t = 0; t < 4; ++t) {
    const int jt = wave * 4 + t;
    const float* Kt = K + ((size_t)(b * SS + jt * 16)) * DM + h * DK;
    v8f acc = {};
    acc = wmma_xt(Qt, DM, Kt, DM, DK, acc);
#pragma unroll
    for (int r = 0; r < 8; ++r)
      sc[r + 8 * half][jt * 16 + l] = acc[r];
  }
  __syncthreads();

  // Phase 2: rel-pos gather + scale + mask + softmax.
  // 8 threads per row, 32 contiguous columns each; reductions within
  // lane-groups of 8 via shuffles.
  {
    const int r = threadIdx.x >> 3;          // row 0..15
    const int g = threadIdx.x & 7;           // column group
    const int i = i0 + r;
    const float* Prow = P + ((size_t)bh * SS + i) * MAXT;
    const int* drow = dists + ((size_t)b * SS + i) * SS;
    const int* mrow = mask + ((size_t)b * SS + i) * SS;  // mask is (B,1,S,S)

    float mx = -3.0e38f;
    for (int jj = 0; jj < 32; ++jj) {
      const int j = g * 32 + jj;
      float s = (sc[r][j] + Prow[drow[j]]) * 0.125f;     // 1/sqrt(64)
      if (mrow[j] == 0) s = -1.0e9f;
      sc[r][j] = s;
      mx = fmaxf(mx, s);
    }
#pragma unroll
    for (int off = 4; off > 0; off >>= 1)
      mx = fmaxf(mx, __shfl_xor(mx, off, 8));

    float sum = 0.0f;
    for (int jj = 0; jj < 32; ++jj) {
      const int j = g * 32 + jj;
      const float e = __expf(sc[r][j] - mx);
      sc[r][j] = e;
      sum += e;
    }
#pragma unroll
    for (int off = 4; off > 0; off >>= 1)
      sum += __shfl_xor(sum, off, 8);

    const float inv = 1.0f / sum;
    for (int jj = 0; jj < 32; ++jj) {
      const int j = g * 32 + jj;
      float p = sc[r][j] * inv;
      if (mrow[j] == 0) p = 0.0f;
      sc[r][j] = p;
    }
  }
  // V panel must be resident before phase 3 reads it.
  async_wait_all();
  __syncthreads();

  // Phase 3: out(16x64) = attn(16x256, LDS) . V_h(256x64, LDS);
  // wave -> d-tile n0. B in "normal" k-major layout from the staged panel.
  {
    const int n0 = wave * 16;
    v8f acc = {};
#pragma unroll 8
    for (int k = 0; k < SS; k += 4) {
      v2f av, bv;
      av.x = sc[l][k + 2 * half];
      av.y = sc[l][k + 2 * half + 1];
      bv.x = Vs[(k + 2 * half) * VS_LDB + n0 + l];
      bv.y = Vs[(k + 2 * half + 1) * VS_LDB + n0 + l];
      acc = __builtin_amdgcn_wmma_f32_16x16x4_f32(false, av, false, bv,
                                                  (short)0, acc, false, false);
    }
    float* xt = X + ((size_t)(b * SS + i0)) * DM + h * DK + n0;
#pragma unroll
    for (int r = 0; r < 8; ++r)
      xt[(size_t)(r + 8 * half) * DM + l] = acc[r];
  }
}

// ---------------------------------------------------------------------------
extern "C" void kernel_launch(void* const* d_in, const int* in_sizes, int n_in,
                              void* d_out, int out_size, void* d_ws, size_t ws_size,
                              hipStream_t stream) {
  (void)in_sizes; (void)n_in; (void)out_size; (void)ws_size;

  const float* query = (const float*)d_in[0];
  const float* key   = (const float*)d_in[1];
  const float* value = (const float*)d_in[2];
  const int*   dists = (const int*)  d_in[3];
  const int*   mask  = (const int*)  d_in[4];
  const float* Wq    = (const float*)d_in[5];
  const float* bq    = (const float*)d_in[6];
  const float* Wk    = (const float*)d_in[7];
  const float* bk    = (const float*)d_in[8];
  const float* Wv    = (const float*)d_in[9];
  const float* bv    = (const float*)d_in[10];
  const float* Wo    = (const float*)d_in[11];
  const float* bo    = (const float*)d_in[12];
  const float* emb   = (const float*)d_in[13];
  float* out = (float*)d_out;

  // workspace: Q,K,V (B*S*D each) + P (B*H*S*MAXT) + X (B*S*D) = 12 MB f32
  float* Qp = (float*)d_ws;
  float* Kp = Qp + (size_t)BB * SS * DM;
  float* Vp = Kp + (size_t)BB * SS * DM;
  float* Pp = Vp + (size_t)BB * SS * DM;
  float* Xp = Pp + (size_t)BB * HH * SS * MAXT;

  const dim3 blk(128);

  // 1) Q/K/V projections: (1024 x 512) = X(1024x512) . W^T + b
  const dim3 gproj(DM / 16, (BB * SS) / 64);
  gemm_xt_bias<<<gproj, blk, 0, stream>>>(query, Wq, bq, Qp, DM, DM, DM);
  gemm_xt_bias<<<gproj, blk, 0, stream>>>(key,   Wk, bk, Kp, DM, DM, DM);
  gemm_xt_bias<<<gproj, blk, 0, stream>>>(value, Wv, bv, Vp, DM, DM, DM);

  // 2) P[b,h] = Q_h (256x64) . emb_h^T (64x128)
  const dim3 gp(MAXT / 16, SS / 64, BB * HH);
  p_gemm<<<gp, blk, 0, stream>>>(Qp, emb, Pp);

  // 3) attention per (b,h,i-block)
  const dim3 ga(SS / 16, BB * HH);
  attention_kernel<<<ga, blk, 0, stream>>>(Qp, Kp, Vp, Pp, dists, mask, Xp);

  // 4) output projection into d_out
  gemm_xt_bias<<<gproj, blk, 0, stream>>>(Xp, Wo, bo, out, DM, DM, DM);
}